// PointFeaturePropagation_10411000726293
// MI455X (gfx1250) — compile-verified
//
#include <hip/hip_runtime.h>
#include <math.h>

// ---------------- problem constants (from reference setup_inputs) ----------
#define BB   4
#define NN   16384
#define MM   4096
#define RR   (BB*NN)        // 65536 rows
#define C1   256            // feat2 channels
#define C2   128            // feat1 channels
#define FINT 128
#define CIN  384            // C1+C2
#define CH   192
#define COUT 128
#define EPS  1e-5f

typedef float v2f __attribute__((ext_vector_type(2)));
typedef float v8f __attribute__((ext_vector_type(8)));

__device__ __forceinline__ float geluf(float x) {
    return 0.5f * x * (1.0f + erff(x * 0.70710678118654752f));
}

// ============================================================================
// Kernel 0: zero the BN-stat accumulators (must run every launch for graphs)
// ============================================================================
__global__ void init_stats_kernel(float* stats) {
    for (int i = threadIdx.x; i < 2048; i += 256) stats[i] = 0.0f;
}

// ============================================================================
// Kernel 1: pack weight matrix [Cin,Cout] into WMMA-B fragment order:
//   Wp[ (k/4)*Cout*4 + n*4 + (k%4) ] = W[k*Cout + n]
// so a lane's B fragment {B[k0][n], B[k0+1][n]} is ONE aligned b64 load.
// ============================================================================
__global__ void pack_w_kernel(const float* __restrict__ W, float* __restrict__ Wp,
                              int Cin, int Cout) {
    int i = blockIdx.x * 256 + threadIdx.x;
    if (i < Cin * Cout) {
        int k = i / Cout, n = i - k * Cout;
        Wp[((size_t)(k >> 2) * Cout + n) * 4 + (k & 3)] = W[i];
    }
}

// ============================================================================
// Kernel 2: three_nn  (one thread per query point, xyz2 batch tile in LDS)
// ============================================================================
__global__ __launch_bounds__(256) void three_nn_kernel(
    const float* __restrict__ xyz1, const float* __restrict__ xyz2,
    int* __restrict__ idx, float* __restrict__ wgt)
{
    __shared__ float sx[MM * 3];                       // 48 KB
    const int blocksPerBatch = NN / 256;
    const int b = blockIdx.x / blocksPerBatch;
    const int qbase = (blockIdx.x % blocksPerBatch) * 256;
    const float* xb = xyz2 + (size_t)b * MM * 3;
    for (int i = threadIdx.x; i < MM * 3; i += 256) sx[i] = xb[i];
    __syncthreads();

    const size_t qi = (size_t)b * NN + qbase + threadIdx.x;
    const float px = xyz1[qi*3+0], py = xyz1[qi*3+1], pz = xyz1[qi*3+2];
    float d0 = 3.4e38f, d1 = 3.4e38f, d2 = 3.4e38f;
    int   i0 = 0, i1 = 0, i2 = 0;
    #pragma unroll 4
    for (int m = 0; m < MM; ++m) {
        float dx = px - sx[3*m+0];
        float dy = py - sx[3*m+1];
        float dz = pz - sx[3*m+2];
        float d  = dx*dx + dy*dy + dz*dz;
        if (d < d2) {
            if (d < d1) {
                d2 = d1; i2 = i1;
                if (d < d0) { d1 = d0; i1 = i0; d0 = d; i0 = m; }
                else        { d1 = d;  i1 = m; }
            } else { d2 = d; i2 = m; }
        }
    }
    float r0 = 1.0f/(d0+1e-8f), r1 = 1.0f/(d1+1e-8f), r2 = 1.0f/(d2+1e-8f);
    float sn = 1.0f/(r0+r1+r2);
    idx[qi*3+0] = i0; idx[qi*3+1] = i1; idx[qi*3+2] = i2;
    wgt[qi*3+0] = r0*sn; wgt[qi*3+1] = r1*sn; wgt[qi*3+2] = r2*sn;
}

// ============================================================================
// Kernel 3: three_interpolate (block = one row, thread = one of C1 channels)
// ============================================================================
__global__ __launch_bounds__(256) void interp_kernel(
    const float* __restrict__ feat2, const int* __restrict__ idx,
    const float* __restrict__ wgt, float* __restrict__ intf)
{
    const size_t r = blockIdx.x;
    const int b = (int)(r >> 14);                      // r / NN
    const int c = threadIdx.x;
    const int*   ip = idx + r*3;
    const float* wp = wgt + r*3;
    const float* f  = feat2 + (size_t)b * MM * C1;
    float v = wp[0]*f[(size_t)ip[0]*C1 + c]
            + wp[1]*f[(size_t)ip[1]*C1 + c]
            + wp[2]*f[(size_t)ip[2]*C1 + c];
    intf[r*C1 + c] = v;
}

// ============================================================================
// Kernel 4: fp32 WMMA GEMM, 32 rows x Cout per block, 8 waves:
//   wave = (rowTile rt = w>>2) x (colGroup cg = w&3), NCT = Cout/64 tiles/wave.
//   A tile staged K-major in LDS with row-stride 40 dwords (bank-disjoint
//   fragment reads for lanes 0-15 vs 16-31); B fragments are single b64
//   loads from the packed weight buffer. K-loop fully unrollable (Cin const).
//   MODE 0: A = A0 raw (+bias)   MODE 1: A = gelu(BN(A0))
//   MODE 2: A = concat(feat1 * sigmoid(BN(psipre)), int_feat)
// ============================================================================
template<int MODE, int Cin, int Cout>
__global__ __launch_bounds__(256) void gemm32_wmma(
    const float* __restrict__ A0, const float* __restrict__ A1,
    const float* __restrict__ Wp, const float* __restrict__ bias,
    float* __restrict__ out,
    const float* __restrict__ sumA, const float* __restrict__ ssA,
    const float* __restrict__ gammaA, const float* __restrict__ betaA,
    const float* __restrict__ psipre, const float* __restrict__ psums,
    const float* __restrict__ gammaP, const float* __restrict__ betaP,
    float Rinv)
{
    constexpr int S   = 40;          // LDS K-major row stride (dwords)
    constexpr int NCT = Cout / 64;   // column tiles per wave
    __shared__ float As[Cin * S];    // <= 384*40*4 = 60 KB
    __shared__ float psiRow[32];

    const int rowBase = blockIdx.x * 32;
    const int t = threadIdx.x;

    if (MODE == 2 && t < 32) {       // per-row sigmoid gate, computed once
        float m   = psums[0] * Rinv;
        float var = psums[1] * Rinv - m * m;
        float ap  = gammaP[0] * rsqrtf(var + EPS);
        float z   = ap * (psipre[rowBase + t] - m) + betaP[0];
        psiRow[t] = 1.0f / (1.0f + expf(-z));
    }
    if (MODE == 2) __syncthreads();

    for (int i = t; i < 32 * Cin; i += 256) {
        const int r = i / Cin;
        const int c = i - r * Cin;
        float v;
        if (MODE == 0) {
            v = A0[(size_t)(rowBase + r) * Cin + c];
        } else if (MODE == 1) {
            float x   = A0[(size_t)(rowBase + r) * Cin + c];
            float m   = sumA[c] * Rinv;
            float var = ssA[c] * Rinv - m * m;
            float a   = gammaA[c] * rsqrtf(var + EPS);
            v = geluf(a * (x - m) + betaA[c]);
        } else {
            if (c < C2) v = A0[(size_t)(rowBase + r) * C2 + c] * psiRow[r];
            else        v = A1[(size_t)(rowBase + r) * C1 + (c - C2)];
        }
        As[c * S + r] = v;
    }
    __syncthreads();

    const int lane  = t & 31;
    const int wave  = t >> 5;
    const int i15   = lane & 15;
    const int k0    = (lane >> 4) << 1;       // 0 or 2
    const int rhalf = (lane >> 4) * 8;        // D rows 0-7 / 8-15
    const int rt    = wave >> 2;              // row tile 0/1
    const int cg    = wave & 3;               // column group 0..3
    const int m32   = rt * 16 + i15;          // A-fragment row within block

    int col[NCT];
    #pragma unroll
    for (int j = 0; j < NCT; ++j) col[j] = (cg * NCT + j) * 16 + i15;

    v8f acc[NCT];
    #pragma unroll
    for (int j = 0; j < NCT; ++j) acc[j] = (v8f){0.f,0.f,0.f,0.f,0.f,0.f,0.f,0.f};

    #pragma unroll 8
    for (int k = 0; k < Cin; k += 4) {
        v2f a;
        a.x = As[(k + k0    ) * S + m32];
        a.y = As[(k + k0 + 1) * S + m32];
        #pragma unroll
        for (int j = 0; j < NCT; ++j) {
            v2f b = *(const v2f*)(Wp + ((size_t)(k >> 2) * Cout + col[j]) * 4 + k0);
            acc[j] = __builtin_amdgcn_wmma_f32_16x16x4_f32(
                         false, a, false, b, (short)0, acc[j], false, false);
        }
    }

    #pragma unroll
    for (int j = 0; j < NCT; ++j) {
        const float bv = bias ? bias[col[j]] : 0.0f;
        #pragma unroll
        for (int i = 0; i < 8; ++i) {
            out[(size_t)(rowBase + rt * 16 + rhalf + i) * Cout + col[j]] = acc[j][i] + bv;
        }
    }
}

// ============================================================================
// Kernel 5: per-channel column sums / sums-of-squares (for BatchNorm stats)
// ============================================================================
__global__ void col_stats_kernel(const float* __restrict__ X,
                                 float* __restrict__ sum, float* __restrict__ ss,
                                 int C, int rows)
{
    const int c = threadIdx.x;
    const size_t r0 = (size_t)blockIdx.x * rows;
    float s = 0.0f, q = 0.0f;
    for (int i = 0; i < rows; ++i) {
        float v = X[(r0 + i) * C + c];
        s += v; q += v * v;
    }
    atomicAdd(&sum[c], s);
    atomicAdd(&ss[c],  q);
}

// ============================================================================
// Kernel 6: attention gate  psipre[r] = gelu(BN(g)+BN(x)) . wp + bp
// ============================================================================
__global__ __launch_bounds__(256) void psi_kernel(
    const float* __restrict__ gpre, const float* __restrict__ xpre,
    const float* __restrict__ sg, const float* __restrict__ qg,
    const float* __restrict__ gg, const float* __restrict__ bgm,
    const float* __restrict__ sx, const float* __restrict__ qx,
    const float* __restrict__ gx, const float* __restrict__ bxm,
    const float* __restrict__ wp, const float* __restrict__ bp,
    float* __restrict__ psipre, float* __restrict__ psums, float Rinv)
{
    __shared__ float red[16];
    const int lane = threadIdx.x & 31;
    const int wave = threadIdx.x >> 5;
    const size_t r = (size_t)blockIdx.x * 8 + wave;

    float acc = 0.0f;
    #pragma unroll
    for (int j = 0; j < 4; ++j) {
        const int c = lane + 32 * j;
        float mg = sg[c] * Rinv, vg = qg[c] * Rinv - mg * mg;
        float ag = gg[c] * rsqrtf(vg + EPS);
        float mx = sx[c] * Rinv, vx = qx[c] * Rinv - mx * mx;
        float ax = gx[c] * rsqrtf(vx + EPS);
        float g = ag * (gpre[r * FINT + c] - mg) + bgm[c];
        float x = ax * (xpre[r * FINT + c] - mx) + bxm[c];
        acc += geluf(g + x) * wp[c];
    }
    #pragma unroll
    for (int o = 16; o > 0; o >>= 1) acc += __shfl_xor(acc, o, 32);
    const float p = acc + bp[0];
    if (lane == 0) { psipre[r] = p; red[wave] = p; red[8 + wave] = p * p; }
    __syncthreads();
    if (threadIdx.x == 0) {
        float s = 0.0f, q = 0.0f;
        for (int w = 0; w < 8; ++w) { s += red[w]; q += red[8 + w]; }
        atomicAdd(&psums[0], s);
        atomicAdd(&psums[1], q);
    }
}

// ============================================================================
// Kernel 7: final  out = gelu(BN(outpre; gamma3, beta3))
// ============================================================================
__global__ __launch_bounds__(256) void bn_gelu_out_kernel(
    const float* __restrict__ pre, const float* __restrict__ s,
    const float* __restrict__ q, const float* __restrict__ g,
    const float* __restrict__ b, float* __restrict__ out, float Rinv)
{
    const size_t i = (size_t)blockIdx.x * blockDim.x + threadIdx.x;
    const int c = (int)(i & (COUT - 1));
    float m   = s[c] * Rinv;
    float var = q[c] * Rinv - m * m;
    float a   = g[c] * rsqrtf(var + EPS);
    out[i] = geluf(a * (pre[i] - m) + b[c]);
}

// ============================================================================
// host-side launch sequence
// ============================================================================
extern "C" void kernel_launch(void* const* d_in, const int* in_sizes, int n_in,
                              void* d_out, int out_size, void* d_ws, size_t ws_size,
                              hipStream_t stream)
{
    (void)in_sizes; (void)n_in; (void)out_size; (void)ws_size;
    const float* xyz1   = (const float*)d_in[0];
    const float* xyz2   = (const float*)d_in[1];
    const float* feat1  = (const float*)d_in[2];
    const float* feat2  = (const float*)d_in[3];
    const float* wg     = (const float*)d_in[4];
    const float* bg     = (const float*)d_in[5];
    const float* gammag = (const float*)d_in[6];
    const float* betag  = (const float*)d_in[7];
    const float* wx     = (const float*)d_in[8];
    const float* bx     = (const float*)d_in[9];
    const float* gammax = (const float*)d_in[10];
    const float* betax  = (const float*)d_in[11];
    const float* wp     = (const float*)d_in[12];
    const float* bp     = (const float*)d_in[13];
    const float* gammap = (const float*)d_in[14];
    const float* betap  = (const float*)d_in[15];
    const float* w1     = (const float*)d_in[16];
    const float* gamma1 = (const float*)d_in[17];
    const float* beta1  = (const float*)d_in[18];
    const float* w2     = (const float*)d_in[19];
    const float* gamma2 = (const float*)d_in[20];
    const float* beta2  = (const float*)d_in[21];
    const float* w3     = (const float*)d_in[22];
    const float* gamma3 = (const float*)d_in[23];
    const float* beta3  = (const float*)d_in[24];
    float* out = (float*)d_out;

    // ---- workspace layout (256B aligned, with aliasing for dead buffers) ----
    char* base = (char*)d_ws;
    size_t cur = 0;
    auto carve = [&](size_t bytes) -> char* {
        char* p = base + cur;
        cur = (cur + bytes + 255) & ~(size_t)255;
        return p;
    };
    int*   ws_idx   = (int*)  carve((size_t)RR * 3 * 4);
    float* ws_wgt   = (float*)carve((size_t)RR * 3 * 4);
    float* ws_intf  = (float*)carve((size_t)RR * C1 * 4);    // 64 MB
    float* ws_gpre  = (float*)carve((size_t)RR * FINT * 4);  // 32 MB
    float* ws_xpre  = (float*)carve((size_t)RR * FINT * 4);  // 32 MB
    float* ws_psip  = (float*)carve((size_t)RR * 4);
    float* ws_h1    = (float*)carve((size_t)RR * CH * 4);    // 48 MB
    float* ws_pwg   = (float*)carve((size_t)C1 * FINT * 4);  // packed weights
    float* ws_pwx   = (float*)carve((size_t)C2 * FINT * 4);
    float* ws_pw1   = (float*)carve((size_t)CIN * CH * 4);
    float* ws_pw2   = (float*)carve((size_t)CH * CH * 4);
    float* ws_pw3   = (float*)carve((size_t)CH * COUT * 4);
    float* ws_stats = (float*)carve(2048 * 4);
    float* ws_h2    = ws_intf;   // int_feat dead after GEMM(w1)
    float* ws_opre  = ws_gpre;   // gpre dead after psi_kernel

    float* sg = ws_stats +    0; float* qg = ws_stats +  128;
    float* sx = ws_stats +  256; float* qx = ws_stats +  384;
    float* ps = ws_stats +  512;                       // 2 scalars
    float* s1 = ws_stats +  640; float* q1 = ws_stats +  832;
    float* s2 = ws_stats + 1024; float* q2 = ws_stats + 1216;
    float* s3 = ws_stats + 1408; float* q3 = ws_stats + 1536;

    const float Rinv = 1.0f / (float)RR;

    // 0) zero BN accumulators (every call -> graph-replay safe)
    init_stats_kernel<<<1, 256, 0, stream>>>(ws_stats);

    // 0b) pack all weight matrices into WMMA-B fragment order
    pack_w_kernel<<<(C1*FINT + 255)/256, 256, 0, stream>>>(wg, ws_pwg, C1, FINT);
    pack_w_kernel<<<(C2*FINT + 255)/256, 256, 0, stream>>>(wx, ws_pwx, C2, FINT);
    pack_w_kernel<<<(CIN*CH  + 255)/256, 256, 0, stream>>>(w1, ws_pw1, CIN, CH);
    pack_w_kernel<<<(CH*CH   + 255)/256, 256, 0, stream>>>(w2, ws_pw2, CH, CH);
    pack_w_kernel<<<(CH*COUT + 255)/256, 256, 0, stream>>>(w3, ws_pw3, CH, COUT);

    // 1) three_nn  -> idx, weight
    three_nn_kernel<<<BB * (NN / 256), 256, 0, stream>>>(xyz1, xyz2, ws_idx, ws_wgt);

    // 2) three_interpolate -> int_feat [RR, 256]
    interp_kernel<<<RR, C1, 0, stream>>>(feat2, ws_idx, ws_wgt, ws_intf);

    // 3) gpre = int_feat @ wg + bg ; xpre = feat1 @ wx + bx   (fp32 WMMA)
    gemm32_wmma<0, C1, FINT><<<RR / 32, 256, 0, stream>>>(
        ws_intf, nullptr, ws_pwg, bg, ws_gpre,
        nullptr, nullptr, nullptr, nullptr, nullptr, nullptr, nullptr, nullptr, Rinv);
    gemm32_wmma<0, C2, FINT><<<RR / 32, 256, 0, stream>>>(
        feat1, nullptr, ws_pwx, bx, ws_xpre,
        nullptr, nullptr, nullptr, nullptr, nullptr, nullptr, nullptr, nullptr, Rinv);

    // 4) BN stats of gpre, xpre
    col_stats_kernel<<<RR / 1024, FINT, 0, stream>>>(ws_gpre, sg, qg, FINT, 1024);
    col_stats_kernel<<<RR / 1024, FINT, 0, stream>>>(ws_xpre, sx, qx, FINT, 1024);

    // 5) psipre = gelu(BN(g)+BN(x)) . wp + bp  (+ scalar stats)
    psi_kernel<<<RR / 8, 256, 0, stream>>>(
        ws_gpre, ws_xpre, sg, qg, gammag, betag, sx, qx, gammax, betax,
        wp, bp, ws_psip, ps, Rinv);

    // 6) h1pre = concat(feat1*psi, int_feat) @ w1
    gemm32_wmma<2, CIN, CH><<<RR / 32, 256, 0, stream>>>(
        feat1, ws_intf, ws_pw1, nullptr, ws_h1,
        nullptr, nullptr, nullptr, nullptr, ws_psip, ps, gammap, betap, Rinv);
    col_stats_kernel<<<RR / 1024, CH, 0, stream>>>(ws_h1, s1, q1, CH, 1024);

    // 7) h2pre = gelu(BN(h1pre; g1,b1)) @ w2
    gemm32_wmma<1, CH, CH><<<RR / 32, 256, 0, stream>>>(
        ws_h1, nullptr, ws_pw2, nullptr, ws_h2,
        s1, q1, gamma1, beta1, nullptr, nullptr, nullptr, nullptr, Rinv);
    col_stats_kernel<<<RR / 1024, CH, 0, stream>>>(ws_h2, s2, q2, CH, 1024);

    // 8) outpre = gelu(BN(h2pre; g2,b2)) @ w3
    gemm32_wmma<1, CH, COUT><<<RR / 32, 256, 0, stream>>>(
        ws_h2, nullptr, ws_pw3, nullptr, ws_opre,
        s2, q2, gamma2, beta2, nullptr, nullptr, nullptr, nullptr, Rinv);
    col_stats_kernel<<<RR / 1024, COUT, 0, stream>>>(ws_opre, s3, q3, COUT, 1024);

    // 9) out = gelu(BN(outpre; g3,b3))
    bn_gelu_out_kernel<<<(RR * COUT) / 256, 256, 0, stream>>>(
        ws_opre, s3, q3, gamma3, beta3, out, Rinv);
}